// GraphQNN_65481071398393
// MI455X (gfx1250) — compile-verified
//
#include <hip/hip_runtime.h>
#include <hip/hip_bf16.h>

// ---------------------------------------------------------------------------
// GraphQNN forward on gfx1250 (CDNA5, wave32).
//   h    = relu(x @ emb_W^T + emb_b)             [20000,256]  (WMMA f32)
//   agg  = scatter_add(h[col] -> row)            [20000,256]  (L2 atomics)
//   hsum = colsum(relu(agg @ msg_W^T + msg_b))   [256]        (WMMA f32,
//          fused epilogue: hm is never materialized)
//   out  = [hsum/N @ W0^T + b0 ; hsum/N @ W1^T + b1]
// ---------------------------------------------------------------------------

typedef __attribute__((ext_vector_type(2))) float v2f;
typedef __attribute__((ext_vector_type(8))) float v8f;

#define GQ_N    20000
#define GQ_NE   640000
#define GQ_HID  256
#define GQ_IN   128

static __device__ __forceinline__ v8f wmma4(v2f a, v2f b, v8f c) {
    return __builtin_amdgcn_wmma_f32_16x16x4_f32(
        /*neg_a=*/false, a, /*neg_b=*/false, b,
        /*c_mod=*/(short)0, c, /*reuse_a=*/false, /*reuse_b=*/false);
}

// relu(A[N,K] @ W[256,K]^T + bias) -> optionally store to O[N,256],
// optionally accumulate column sums into csum[256].
// Grid: (N/16) blocks, 256 threads = 8 waves. Wave w owns 16 rows x 32 cols
// (two 16x16 tiles sharing one A fragment -> two independent WMMA chains).
__global__ __launch_bounds__(256)
void gemm_relu_wmma(const float* __restrict__ A, const float* __restrict__ W,
                    const float* __restrict__ bias, float* __restrict__ O,
                    float* __restrict__ csum, int K) {
    const int lane = threadIdx.x & 31;
    const int wave = threadIdx.x >> 5;
    const int row0 = blockIdx.x * 16;
    const int col0 = wave * 32;    // 8 waves cover all 256 cols
    const int m  = lane & 15;      // M for A, N for B
    const int kq = lane >> 4;      // lanes 16-31 hold K=+2,+3

    // 16x4 f32 A: lane holds (M=m, K=k0+kq*2..+1) -> contiguous float2
    const float* arow  = A + (size_t)(row0 + m) * K + kq * 2;
    // 4x16 f32 B: B[k][n] = W[col+n][k]; lane holds (N=m, same K pair)
    const float* brow0 = W + (size_t)(col0 + m) * K + kq * 2;
    const float* brow1 = brow0 + (size_t)16 * K;

    v8f acc0 = {};
    v8f acc1 = {};
#pragma unroll 4
    for (int k0 = 0; k0 < K; k0 += 4) {
        v2f av  = *(const v2f*)(arow  + k0);
        v2f bv0 = *(const v2f*)(brow0 + k0);
        v2f bv1 = *(const v2f*)(brow1 + k0);
        acc0 = wmma4(av, bv0, acc0);   // two independent accumulation chains
        acc1 = wmma4(av, bv1, acc1);
    }

    // C/D layout: lane L, VGPR r -> (M = r + (L/16)*8, N = L%16)
    const int n  = lane & 15;
    const int r0 = kq * 8;
    const float bn0 = bias[col0 + n];
    const float bn1 = bias[col0 + 16 + n];
    float s0 = 0.0f, s1 = 0.0f;
#pragma unroll
    for (int r = 0; r < 8; ++r) {
        float v0 = fmaxf(acc0[r] + bn0, 0.0f);
        float v1 = fmaxf(acc1[r] + bn1, 0.0f);
        s0 += v0;
        s1 += v1;
        if (O) {
            const size_t ro = (size_t)(row0 + r0 + r) * GQ_HID;
            O[ro + col0 + n]      = v0;
            O[ro + col0 + 16 + n] = v1;
        }
    }
    if (csum) {
        // combine the two 8-row halves (lane <-> lane^16 hold same column)
        s0 += __shfl_xor(s0, 16, 32);
        s1 += __shfl_xor(s1, 16, 32);
        if (lane < 16) {
            atomicAdd(csum + col0 + n,      s0);
            atomicAdd(csum + col0 + 16 + n, s1);
        }
    }
}

// agg[dst,:] += h[src,:] per edge; thread = channel (coalesced), 4 edges/block.
// h/agg are 20MB each -> gathers and f32 atomics resolve in the 192MB L2.
__global__ __launch_bounds__(256)
void scatter_add_kernel(const float* __restrict__ H, const int* __restrict__ edges,
                        float* __restrict__ AGG, int nE) {
    const int c  = threadIdx.x;
    const int e0 = blockIdx.x * 4;
#pragma unroll
    for (int r = 0; r < 4; ++r) {
        const int e = e0 + r;
        if (e < nE) {
            const int dst = edges[e];        // edge_index[0]: destinations
            const int src = edges[nE + e];   // edge_index[1]: sources
            atomicAdd(AGG + (size_t)dst * GQ_HID + c, H[(size_t)src * GQ_HID + c]);
        }
    }
}

__global__ __launch_bounds__(256)
void zero_kernel(float* __restrict__ p, int n) {
    const int i = blockIdx.x * blockDim.x + threadIdx.x;
    if (i < n) p[i] = 0.0f;
}

// out[j] = b[j] + (hsum/N) . W[j,:]   for W0 (j<8192) then W1 (j>=8192)
__global__ __launch_bounds__(256)
void head_kernel(const float* __restrict__ hsum,
                 const float* __restrict__ W0, const float* __restrict__ b0,
                 const float* __restrict__ W1, const float* __restrict__ b1,
                 float* __restrict__ out, float invN) {
    __shared__ float4 hs[GQ_HID / 4];
    const int t = threadIdx.x;
    if (t < GQ_HID / 4) {
        float4 v = ((const float4*)hsum)[t];
        v.x *= invN; v.y *= invN; v.z *= invN; v.w *= invN;
        hs[t] = v;
    }
    __syncthreads();

    const int j = blockIdx.x * 256 + t;      // 0..10239
    const float4* Wr;
    float bias;
    if (j < GQ_IN * 64) {                    // 8192 -> w0
        Wr = (const float4*)(W0 + (size_t)j * GQ_HID);
        bias = b0[j];
    } else {                                 // 2048 -> w1
        const int jj = j - GQ_IN * 64;
        Wr = (const float4*)(W1 + (size_t)jj * GQ_HID);
        bias = b1[jj];
    }
    float s = bias;
#pragma unroll
    for (int k = 0; k < GQ_HID / 4; ++k) {
        const float4 w = Wr[k];
        const float4 h = hs[k];
        s += w.x * h.x + w.y * h.y + w.z * h.z + w.w * h.w;
    }
    out[j] = s;
}

extern "C" void kernel_launch(void* const* d_in, const int* in_sizes, int n_in,
                              void* d_out, int out_size, void* d_ws, size_t ws_size,
                              hipStream_t stream) {
    const float* x     = (const float*)d_in[0];
    const int*   edges = (const int*)d_in[1];
    const float* emb_W = (const float*)d_in[2];
    const float* emb_b = (const float*)d_in[3];
    const float* msg_W = (const float*)d_in[4];
    const float* msg_b = (const float*)d_in[5];
    const float* W0    = (const float*)d_in[6];
    const float* b0    = (const float*)d_in[7];
    const float* W1    = (const float*)d_in[8];
    const float* b1    = (const float*)d_in[9];
    float* out = (float*)d_out;

    // workspace: h | agg | hsum  (~41 MB)
    float* h    = (float*)d_ws;
    float* agg  = h   + (size_t)GQ_N * GQ_HID;
    float* hsum = agg + (size_t)GQ_N * GQ_HID;

    // 1) zero accumulators
    zero_kernel<<<(GQ_N * GQ_HID + 255) / 256, 256, 0, stream>>>(agg, GQ_N * GQ_HID);
    zero_kernel<<<1, 256, 0, stream>>>(hsum, GQ_HID);

    // 2) h = relu(x @ emb_W^T + emb_b)        [K=128], store h
    gemm_relu_wmma<<<GQ_N / 16, 256, 0, stream>>>(
        x, emb_W, emb_b, h, nullptr, GQ_IN);

    // 3) agg[row] += h[col] over edges (roofline-dominant stage)
    scatter_add_kernel<<<GQ_NE / 4, 256, 0, stream>>>(h, edges, agg, GQ_NE);

    // 4) hsum = colsum(relu(agg @ msg_W^T + msg_b))  [K=256], no store
    gemm_relu_wmma<<<GQ_N / 16, 256, 0, stream>>>(
        agg, msg_W, msg_b, nullptr, hsum, GQ_HID);

    // 5) out = [hsum/N @ W0^T + b0 ; hsum/N @ W1^T + b1]
    head_kernel<<<(GQ_IN * 64 + 64 * 32) / 256, 256, 0, stream>>>(
        hsum, W0, b0, W1, b1, out, 1.0f / (float)GQ_N);
}